// LocalAttention_50483045597573
// MI455X (gfx1250) — compile-verified
//
#include <hip/hip_runtime.h>
#include <math.h>

#if defined(__HIP_DEVICE_COMPILE__)
#if !__has_builtin(__builtin_amdgcn_wmma_f32_16x16x4_f32)
#error "gfx1250 device pass: __builtin_amdgcn_wmma_f32_16x16x4_f32 not available"
#endif
#endif

typedef float v2f __attribute__((ext_vector_type(2)));
typedef float v8f __attribute__((ext_vector_type(8)));

constexpr int WINW = 128;   // window size
constexpr int DH   = 64;    // head dim
constexpr int NSEQ = 8192;  // sequence length
constexpr int QSTR = 68;    // padded LDS row stride (floats): bank-conflict free, 16B aligned
constexpr int PSTR = 18;    // P transpose tile stride
#define NEG_MAX -3.402823466e38f

__device__ __forceinline__ v8f wmma4(v2f a, v2f b, v8f c) {
#if defined(__HIP_DEVICE_COMPILE__)
  // D = A(16x4,f32) * B(4x16,f32) + C(16x16,f32)
  return __builtin_amdgcn_wmma_f32_16x16x4_f32(false, a, false, b, (short)0, c, false, false);
#else
  return c;
#endif
}

__device__ __forceinline__ float rmax16(float v) {
#pragma unroll
  for (int m = 1; m < 16; m <<= 1) v = fmaxf(v, __shfl_xor(v, m, 32));
  return v;
}
__device__ __forceinline__ float rsum16(float v) {
#pragma unroll
  for (int m = 1; m < 16; m <<= 1) v += __shfl_xor(v, m, 32);
  return v;
}

// Load one 64-wide row from global, apply RoPE (local position t) and scale, store to LDS.
__device__ __forceinline__ void rope_row(const float* __restrict__ src,
                                         float* __restrict__ dst,
                                         float t, float scale) {
  const float LG = 0.41524101186092029f; // log2(10000)/32
#pragma unroll
  for (int i = 0; i < 8; ++i) {
    const int d0 = i * 4;
    float4 x1 = *(const float4*)(src + d0);
    float4 x2 = *(const float4*)(src + d0 + 32);
    const float* p1 = (const float*)&x1;
    const float* p2 = (const float*)&x2;
    float o1[4], o2[4];
#pragma unroll
    for (int u = 0; u < 4; ++u) {
      float f = t * exp2f(-(float)(d0 + u) * LG);
      float s, c;
      __sincosf(f, &s, &c);
      float a = p1[u] * scale;
      float b = p2[u] * scale;
      o1[u] = a * c - b * s;   // d  : x*cos - x[d+32]*sin
      o2[u] = b * c + a * s;   // d+32: x*cos + x[d]*sin
    }
    *(float4*)(dst + d0)      = make_float4(o1[0], o1[1], o1[2], o1[3]);
    *(float4*)(dst + d0 + 32) = make_float4(o2[0], o2[1], o2[2], o2[3]);
  }
}

__global__ __launch_bounds__(256)
void local_attn_f32wmma(const float* __restrict__ q,
                        const float* __restrict__ k,
                        const float* __restrict__ v,
                        float* __restrict__ out) {
  __shared__ float Qs[WINW * QSTR];        // RoPE'd, scaled queries
  __shared__ float Ks[WINW * QSTR];        // RoPE'd keys, one 128-row half at a time
  __shared__ float Vs[WINW * QSTR];        // values, one half at a time
  __shared__ float Ps[8 * 16 * PSTR];      // per-wave P-transpose staging

  const int win  = blockIdx.x;
  const int bh   = blockIdx.y;
  const int tid  = threadIdx.x;
  const int lane = tid & 31;
  const int wid  = tid >> 5;     // wave 0..7 -> query rows [16*wid, 16*wid+15]
  const int l16  = lane & 15;
  const int hh   = lane >> 4;    // lane half
  const int mbase = wid * 16;

  const size_t base = (size_t)bh * NSEQ * DH;
  const int firsthalf = (win == 0) ? 1 : 0;   // window 0: previous window is all padding

  // ---- initial loads: Q (threads 0..127), first K half (threads 128..255), V half (all) ----
  if (tid < WINW) {
    rope_row(q + base + (size_t)(win * WINW + tid) * DH,
             Qs + tid * QSTR, (float)(WINW + tid), 0.125f);   // scale = D^-0.5
  } else {
    int jj = tid - WINW;
    int kr = (win - 1 + firsthalf) * WINW + jj;
    rope_row(k + base + (size_t)kr * DH, Ks + jj * QSTR,
             (float)(firsthalf * WINW + jj), 1.0f);
  }
  {
    int row = tid >> 1, dh = (tid & 1) * 32;
    int vr = (win - 1 + firsthalf) * WINW + row;
    const float4* s = (const float4*)(v + base + (size_t)vr * DH + dh);
    float4* dd = (float4*)(Vs + row * QSTR + dh);
#pragma unroll
    for (int i = 0; i < 8; ++i) dd[i] = s[i];
  }
  __syncthreads();

  // Preload this wave's 16 Q A-fragments (K-slices of 4): lane<16 holds K=4kk+{0,1}, lane>=16 K=4kk+{2,3}
  v2f qa[16];
#pragma unroll
  for (int kk = 0; kk < 16; ++kk) {
    int a = (mbase + l16) * QSTR + 4 * kk + 2 * hh;
    qa[kk].x = Qs[a];
    qa[kk].y = Qs[a + 1];
  }

  float mrun[8], lrun[8];
  v8f acc[4];
#pragma unroll
  for (int r = 0; r < 8; ++r) { mrun[r] = NEG_MAX; lrun[r] = 0.f; }
#pragma unroll
  for (int t = 0; t < 4; ++t) { v8f z = {}; acc[t] = z; }

  float* pw = Ps + wid * 16 * PSTR;

  for (int h = firsthalf; h < 2; ++h) {
    if (h != firsthalf) {
      __syncthreads();               // all waves done with previous K/V half
      if (tid < WINW) {
        int kr = win * WINW + tid;   // h == 1 -> current window keys
        rope_row(k + base + (size_t)kr * DH, Ks + tid * QSTR,
                 (float)(WINW + tid), 1.0f);
      }
      {
        int row = tid >> 1, dh = (tid & 1) * 32;
        int vr = win * WINW + row;
        const float4* s = (const float4*)(v + base + (size_t)vr * DH + dh);
        float4* dd = (float4*)(Vs + row * QSTR + dh);
#pragma unroll
        for (int i = 0; i < 8; ++i) dd[i] = s[i];
      }
      __syncthreads();
    }

    // half 0: all 8 chunks, never masked. half 1: wave w needs chunks 0..w, diagonal chunk is triangular.
    const int nch  = (h == 0) ? 8 : (wid + 1);
    const int diag = (h == 0) ? -1 : wid;

    for (int c = 0; c < nch; ++c) {
      const int jc = c * 16;

      // ---- S tile = Q(16x64) * K^T(64x16), two interleaved WMMA chains ----
      v8f s0 = {}, s1 = {};
#pragma unroll
      for (int kk = 0; kk < 16; kk += 2) {
        const int rb = (jc + l16) * QSTR + 2 * hh;
        v2f b0, b1;
        b0.x = Ks[rb + 4 * kk];           b0.y = Ks[rb + 4 * kk + 1];
        b1.x = Ks[rb + 4 * (kk + 1)];     b1.y = Ks[rb + 4 * (kk + 1) + 1];
        s0 = wmma4(qa[kk],     b0, s0);
        s1 = wmma4(qa[kk + 1], b1, s1);
      }

      const bool dc = (c == diag);
      float sarr[8];
#pragma unroll
      for (int r = 0; r < 8; ++r) {
        float sv = s0[r] + s1[r];
        if (dc && (l16 > r + 8 * hh)) sv = NEG_MAX;   // in-tile causal triangle
        sarr[r] = sv;
      }

      // ---- online softmax (row = r + 8*hh, cols across the 16-lane group) ----
      float p[8];
#pragma unroll
      for (int r = 0; r < 8; ++r) {
        float cm    = rmax16(sarr[r]);
        float mnew  = fmaxf(mrun[r], cm);
        float alpha = __expf(mrun[r] - mnew);
        mrun[r] = mnew;
        float pv = __expf(sarr[r] - mnew);
        if (dc && (l16 > r + 8 * hh)) pv = 0.f;
        p[r] = pv;
        lrun[r] = lrun[r] * alpha + rsum16(pv);
#pragma unroll
        for (int t = 0; t < 4; ++t) acc[t][r] *= alpha;
      }

      // ---- transpose P (C layout -> A layout) through per-wave LDS tile ----
#pragma unroll
      for (int r = 0; r < 8; ++r)
        pw[(r + 8 * hh) * PSTR + l16] = p[r];
      asm volatile("s_wait_dscnt 0" ::: "memory");   // same-wave store->load ordering

      // ---- O += P(16x16) * V(16x64) ----
#pragma unroll
      for (int s = 0; s < 4; ++s) {
        v2f a;
        const int ab = l16 * PSTR + 4 * s + 2 * hh;
        a.x = pw[ab]; a.y = pw[ab + 1];
#pragma unroll
        for (int t = 0; t < 4; ++t) {
          v2f b;
          const int vb = (jc + 4 * s + 2 * hh) * QSTR + 16 * t + l16;
          b.x = Vs[vb]; b.y = Vs[vb + QSTR];
          acc[t] = wmma4(a, b, acc[t]);
        }
      }
    }
  }

  // ---- normalize and store ----
  float* ob = out + base + (size_t)win * WINW * DH;
#pragma unroll
  for (int r = 0; r < 8; ++r) {
    float inv = 1.0f / lrun[r];
    int row = mbase + r + 8 * hh;
#pragma unroll
    for (int t = 0; t < 4; ++t)
      ob[row * DH + 16 * t + l16] = acc[t][r] * inv;
  }
}

extern "C" void kernel_launch(void* const* d_in, const int* in_sizes, int n_in,
                              void* d_out, int out_size, void* d_ws, size_t ws_size,
                              hipStream_t stream) {
  const float* q = (const float*)d_in[0];
  const float* k = (const float*)d_in[1];
  const float* v = (const float*)d_in[2];
  float* out = (float*)d_out;
  const int BH = in_sizes[0] / (NSEQ * DH);     // 64
  dim3 grid(NSEQ / WINW, BH);                   // (64 windows, 64 batch*heads)
  local_attn_f32wmma<<<grid, 256, 0, stream>>>(q, k, v, out);
}